// MyHMM_10118942949539
// MI455X (gfx1250) — compile-verified
//
#include <hip/hip_runtime.h>
#include <hip/hip_bf16.h>

#define B_  64
#define T_  8192
#define C_  16
#define XD_ 6
#define LOG2PI_F   1.8378770664093453f
#define LN2_F      0.6931471805599453f
#define INV_LN2_F  1.4426950408889634f
#define HALF_INV_LN2_F 0.7213475204444817f

typedef __attribute__((ext_vector_type(2))) float v2f;
typedef __attribute__((ext_vector_type(8))) float v8f;

// ---- workspace layout (bytes) ----
#define WS_LOGA   0u                       // 256 f  (log2-softmax rows of A)
#define WS_EXPA   1024u                    // 256 f  (softmax(A) probabilities)
#define WS_LOGPI  2048u                    // 16 f   (log2-softmax pi)
#define WS_MU     2112u                    // 64 f   (means[:, :4])
#define WS_INVS   2368u                    // 64 f   (1/sigma)
#define WS_CONST  2624u                    // 16 f   (log2-units emission const)
#define WS_CLAST  2688u                    // 64 int
#define WS_ELP    4096u                    // B*T*C floats, layout [g][t][b%16][c], log2 units
#define WS_PSI    (WS_ELP + (size_t)B_*T_*C_*4)   // B*T*C bytes

// ---------------------------------------------------------------------------
// Pass 0: normalize logA/logpi (log2 units), precompute emission constants.
// ---------------------------------------------------------------------------
__global__ void hmm_prep(const float* __restrict__ means,
                         const float* __restrict__ stds,
                         const float* __restrict__ logA_in,
                         const float* __restrict__ logpi_in,
                         char* __restrict__ ws) {
    float* wLogA  = (float*)(ws + WS_LOGA);
    float* wExpA  = (float*)(ws + WS_EXPA);
    float* wLogPi = (float*)(ws + WS_LOGPI);
    float* wMu    = (float*)(ws + WS_MU);
    float* wInvS  = (float*)(ws + WS_INVS);
    float* wConst = (float*)(ws + WS_CONST);

    int t = threadIdx.x;
    if (t < 16) {
        // row t log_softmax of log_A (natural), stored in log2 units
        float row[16], mx = -3.4e38f;
        for (int j = 0; j < 16; ++j) { row[j] = logA_in[t * 16 + j]; mx = fmaxf(mx, row[j]); }
        float ss = 0.f;
        for (int j = 0; j < 16; ++j) ss += __expf(row[j] - mx);
        float lse = mx + __logf(ss);
        for (int j = 0; j < 16; ++j) {
            float v = row[j] - lse;                 // natural-log softmax
            wLogA[t * 16 + j] = v * INV_LN2_F;      // log2 units
            wExpA[t * 16 + j] = __expf(v);          // probability
        }
        // emission constants for state t (log2 units)
        float cc = -2.0f * LOG2PI_F;                // 4 dims * 0.5*log2pi
        for (int d = 0; d < 4; ++d) {
            float sg = fmaxf(stds[t * XD_ + d], 0.f) + 0.1f;   // relu + 0.1
            cc -= __logf(sg);
            wMu[t * 4 + d]   = means[t * XD_ + d];
            wInvS[t * 4 + d] = 1.0f / sg;
        }
        wConst[t] = cc * INV_LN2_F;
    } else if (t == 16) {
        float row[16], mx = -3.4e38f;
        for (int j = 0; j < 16; ++j) { row[j] = logpi_in[j]; mx = fmaxf(mx, row[j]); }
        float ss = 0.f;
        for (int j = 0; j < 16; ++j) ss += __expf(row[j] - mx);
        float lse = mx + __logf(ss);
        for (int j = 0; j < 16; ++j) wLogPi[j] = (row[j] - lse) * INV_LN2_F;
    }
}

// ---------------------------------------------------------------------------
// Pass 1: emission log2-probs. One thread per (b,t); 16 contiguous floats
// (64B aligned -> global_store_b128). Layout: elp[g][t][b_local][c].
// ---------------------------------------------------------------------------
__global__ __launch_bounds__(256) void hmm_emit(const float* __restrict__ x,
                                                char* __restrict__ ws) {
    int idx = blockIdx.x * 256 + threadIdx.x;           // exactly B*T threads
    int b = idx / T_;
    int t = idx % T_;
    const float* xp = x + ((size_t)b * T_ + t) * XD_;
    float x0 = xp[0], x1 = xp[1], x2 = xp[2], x3 = xp[3];

    const float* mu  = (const float*)(ws + WS_MU);
    const float* is_ = (const float*)(ws + WS_INVS);
    const float* cst = (const float*)(ws + WS_CONST);

    float* ep = (float*)(ws + WS_ELP) + ((size_t)(b >> 4) * T_ + t) * 256 + (b & 15) * 16;
#pragma unroll
    for (int c = 0; c < 16; ++c) {
        float z0 = (x0 - mu[c * 4 + 0]) * is_[c * 4 + 0];
        float z1 = (x1 - mu[c * 4 + 1]) * is_[c * 4 + 1];
        float z2 = (x2 - mu[c * 4 + 2]) * is_[c * 4 + 2];
        float z3 = (x3 - mu[c * 4 + 3]) * is_[c * 4 + 3];
        ep[c] = cst[c] - (z0 * z0 + z1 * z1 + z2 * z2 + z3 * z3) * HALF_INV_LN2_F;
    }
}

// ---------------------------------------------------------------------------
// Pass 2: block 0 = forward scan (WMMA f32 16x16x4, 16 batches/wave, 4 waves,
//                  no block barriers: all LDS traffic is intra-wave)
//         blocks 1..8 = Viterbi (2 batches/wave, state = lane&15)
// ---------------------------------------------------------------------------
__global__ __launch_bounds__(128) void hmm_fwd_vit(char* __restrict__ ws,
                                                   float* __restrict__ out) {
    const float* logA  = (const float*)(ws + WS_LOGA);
    const float* expA  = (const float*)(ws + WS_EXPA);
    const float* logpi = (const float*)(ws + WS_LOGPI);
    const float* elp   = (const float*)(ws + WS_ELP);

    // per-wave slab: holds raw scores (log2) transiently, then P = exp2(norm)
    __shared__ float pS[4][16][17];

    int tid  = threadIdx.x;
    int w    = tid >> 5;                // wave id in block
    int lane = tid & 31;
    int h    = lane >> 4;               // half-wave (0/1)
    int ln   = lane & 15;

    if (blockIdx.x == 0) {
        // ------------------- forward (scaled, prob-space, WMMA) -------------
        const float* elpG = elp + (size_t)w * T_ * 256;   // this wave's 16 batches

        // B-operand: softmax(A) in WMMA 4x16 f32 B layout (constant all steps)
        v2f bOp[4];
#pragma unroll
        for (int s = 0; s < 4; ++s) {
            bOp[s].x = expA[(4 * s + 2 * h + 0) * 16 + ln];
            bOp[s].y = expA[(4 * s + 2 * h + 1) * 16 + ln];
        }

        // t = 0 : raw score = elp2 + logpi2
#pragma unroll
        for (int k = 0; k < 8; ++k) {
            int idx = lane + 32 * k;
            pS[w][idx >> 4][idx & 15] = elpG[idx] + logpi[idx & 15];
        }
        __builtin_amdgcn_wave_barrier();

        float slog = 0.f;                // sum of per-step log2-scalers
        // normalize raw scores -> P in LDS, accumulate scaler (lanes 0..15)
        {
            if (lane < 16) {
                float a[16], mx = -3.4e38f;
#pragma unroll
                for (int j = 0; j < 16; ++j) { a[j] = pS[w][lane][j]; mx = fmaxf(mx, a[j]); }
                float e[16], ss = 0.f;
#pragma unroll
                for (int j = 0; j < 16; ++j) { e[j] = __builtin_amdgcn_exp2f(a[j] - mx); ss += e[j]; }
                float rs = 1.0f / ss;
#pragma unroll
                for (int j = 0; j < 16; ++j) pS[w][lane][j] = e[j] * rs;
                slog += mx + __builtin_amdgcn_logf(ss);
            }
            __builtin_amdgcn_wave_barrier();
        }

        for (int t = 1; t < T_; ++t) {
            const float* et = elpG + (size_t)t * 256;
            __builtin_prefetch(et + 256 + lane * 8, 0, 1);   // next step's block

            // A-operand: P (already probabilities), 16x4 slices (M=batch,K=state)
            v2f aOp[4];
#pragma unroll
            for (int s = 0; s < 4; ++s) {
                aOp[s].x = pS[w][ln][4 * s + 2 * h + 0];
                aOp[s].y = pS[w][ln][4 * s + 2 * h + 1];
            }
            __builtin_amdgcn_wave_barrier();

            // D[m][j] = sum_k P[m][k] * A[k][j], K=16 via 4 chained WMMAs
            v8f acc = {};
#pragma unroll
            for (int s = 0; s < 4; ++s)
                acc = __builtin_amdgcn_wmma_f32_16x16x4_f32(
                        false, aOp[s], false, bOp[s], (short)0, acc, false, false);

            // raw score = log2(D) + elp2; stash in LDS for renormalize
#pragma unroll
            for (int v = 0; v < 8; ++v) {
                int m2 = v + 8 * h;                       // C/D layout row
                pS[w][m2][ln] = __builtin_amdgcn_logf(acc[v]) + et[m2 * 16 + ln];
            }
            __builtin_amdgcn_wave_barrier();

            if (lane < 16) {
                float a[16], mx = -3.4e38f;
#pragma unroll
                for (int j = 0; j < 16; ++j) { a[j] = pS[w][lane][j]; mx = fmaxf(mx, a[j]); }
                float e[16], ss = 0.f;
#pragma unroll
                for (int j = 0; j < 16; ++j) { e[j] = __builtin_amdgcn_exp2f(a[j] - mx); ss += e[j]; }
                float rs = 1.0f / ss;
#pragma unroll
                for (int j = 0; j < 16; ++j) pS[w][lane][j] = e[j] * rs;
                slog += mx + __builtin_amdgcn_logf(ss);
            }
            __builtin_amdgcn_wave_barrier();
        }
        if (lane < 16) out[w * 16 + lane] = slog * LN2_F;   // logp_x (natural log)
    } else {
        // --------------------------- Viterbi (log2 units, order-invariant) ---
        int b  = (blockIdx.x - 1) * 8 + w * 2 + h;         // 2 batches per wave
        int j  = ln;                                       // state = lane&15
        int g  = b >> 4, bl = b & 15;
        const float* elpB = elp + (size_t)g * T_ * 256 + bl * 16 + j;
        unsigned char* psi = (unsigned char*)(ws + WS_PSI) + (size_t)b * T_ * 16 + j;

        float colA[16];
#pragma unroll
        for (int i = 0; i < 16; ++i) colA[i] = logA[i * 16 + j];

        float delta = elpB[0] + logpi[j];
        int srcBase = lane & 0x10;
        for (int t = 1; t < T_; ++t) {
            __builtin_prefetch(elpB + (size_t)(t + 1) * 256, 0, 1);
            float best = -3.4e38f; int bi = 0;
#pragma unroll
            for (int i = 0; i < 16; ++i) {
                float dv = __shfl(delta, srcBase + i, 32);
                float cand = dv + colA[i];
                if (cand > best) { best = cand; bi = i; }   // first-max tiebreak
            }
            delta = best + elpB[(size_t)t * 256];
            psi[(size_t)t * 16] = (unsigned char)bi;
        }
        // argmax over states within the 16-lane group (lowest index on tie)
        float bv = delta; int bidx = j;
#pragma unroll
        for (int off = 8; off >= 1; off >>= 1) {
            float ov = __shfl_xor(bv, off, 32);
            int   oi = __shfl_xor(bidx, off, 32);
            if (ov > bv || (ov == bv && oi < bidx)) { bv = ov; bidx = oi; }
        }
        if (j == 0) ((int*)(ws + WS_CLAST))[b] = bidx;
    }
}

// ---------------------------------------------------------------------------
// Pass 3: backtrace, one thread per batch.
// ---------------------------------------------------------------------------
__global__ void hmm_backtrace(const unsigned char* __restrict__ psi,
                              const int* __restrict__ cLast,
                              float* __restrict__ out) {
    int b = threadIdx.x;
    if (b >= B_) return;
    int c = cLast[b];
    float* co = out + B_ + (size_t)b * T_;
    co[T_ - 1] = (float)c;
    for (int t = T_ - 2; t >= 0; --t) {
        c = psi[((size_t)b * T_ + (t + 1)) * 16 + c];
        co[t] = (float)c;
    }
}

extern "C" void kernel_launch(void* const* d_in, const int* in_sizes, int n_in,
                              void* d_out, int out_size, void* d_ws, size_t ws_size,
                              hipStream_t stream) {
    const float* x     = (const float*)d_in[0];   // (B,T,6)
    const float* means = (const float*)d_in[1];   // (16,6)
    const float* stds  = (const float*)d_in[2];   // (16,6)
    const float* logA  = (const float*)d_in[3];   // (16,16)
    const float* logpi = (const float*)d_in[4];   // (16,)
    char*  ws  = (char*)d_ws;
    float* out = (float*)d_out;                   // [64 logp_x][B*T c_est]

    hmm_prep<<<1, 32, 0, stream>>>(means, stds, logA, logpi, ws);
    hmm_emit<<<(B_ * T_) / 256, 256, 0, stream>>>(x, ws);
    hmm_fwd_vit<<<9, 128, 0, stream>>>(ws, out);
    hmm_backtrace<<<1, 64, 0, stream>>>((const unsigned char*)(ws + WS_PSI),
                                        (const int*)(ws + WS_CLAST), out);
}